// GCN_15333033247254
// MI455X (gfx1250) — compile-verified
//
#include <hip/hip_runtime.h>
#include <math.h>

typedef __attribute__((ext_vector_type(2))) float v2f;
typedef __attribute__((ext_vector_type(8))) float v8f;

__device__ __forceinline__ void atomAddF(float* p, float v) {
  __hip_atomic_fetch_add(p, v, __ATOMIC_RELAXED, __HIP_MEMORY_SCOPE_AGENT);
}

// ---------------- graph normalization ----------------
__global__ void k_deg_init(float* deg, int N) {
  int i = blockIdx.x * blockDim.x + threadIdx.x;
  if (i < N) deg[i] = 1.0f;                 // self-loop contribution
}

__global__ void k_deg_edges(const long long* __restrict__ ei, float* deg, int E) {
  int e = blockIdx.x * blockDim.x + threadIdx.x;
  if (e < E) atomAddF(&deg[(int)ei[(long)E + e]], 1.0f);
}

__global__ void k_rsqrt_inplace(float* deg, int N) {
  int i = blockIdx.x * blockDim.x + threadIdx.x;
  if (i < N) deg[i] = rsqrtf(deg[i]);       // deg >= 1 always (self-loop)
}

__global__ void k_edge_norm(const long long* __restrict__ ei,
                            const float* __restrict__ dis,
                            float* __restrict__ nrm, int E) {
  int e = blockIdx.x * blockDim.x + threadIdx.x;
  if (e < E) nrm[e] = dis[(int)ei[e]] * dis[(int)ei[(long)E + e]];
}

// ---------------- WMMA GEMM1: [N,128] x [128,128], fused epilogue ------------
// One wave computes a 16-row strip across all 128 output columns
// (8 accumulators), so X is streamed exactly once. Epilogue stores the raw
// product (for the edge gather) AND the aggregation init
// hbuf = b1 + xw * dis^2 (bias + self-loop message), saving a full pass.
__global__ void k_gemm1_wmma(const float* __restrict__ X,
                             const float* __restrict__ W,
                             const float* __restrict__ dis,
                             const float* __restrict__ bias,
                             float* __restrict__ XW,
                             float* __restrict__ HB, int nStrips) {
  int wave = (blockIdx.x * blockDim.x + threadIdx.x) >> 5;
  if (wave >= nStrips) return;              // wave-uniform: EXEC all-1s inside
  int lane = threadIdx.x & 31;
  int rc = lane & 15;                       // row for A, col for B/D
  int h  = lane >> 4;
  long abase = (long)(wave * 16 + rc) * 128;

  v8f acc[8];
#pragma unroll
  for (int nt = 0; nt < 8; ++nt) acc[nt] = (v8f){0,0,0,0,0,0,0,0};

  for (int k0 = 0; k0 < 128; k0 += 4) {
    float2 av = *(const float2*)(X + abase + k0 + 2 * h);
    v2f a; a[0] = av.x; a[1] = av.y;
#pragma unroll
    for (int nt = 0; nt < 8; ++nt) {
      v2f b;
      b[0] = W[(k0 + 2 * h)     * 128 + nt * 16 + rc];
      b[1] = W[(k0 + 2 * h + 1) * 128 + nt * 16 + rc];
      acc[nt] = __builtin_amdgcn_wmma_f32_16x16x4_f32(
          false, a, false, b, (short)0, acc[nt], false, false);
    }
  }

  // D layout: lane holds col rc; element r is row (r + 8h) of the strip
  float dsq[8];
#pragma unroll
  for (int r = 0; r < 8; ++r) {
    float t = dis[wave * 16 + r + 8 * h];
    dsq[r] = t * t;
  }
#pragma unroll
  for (int nt = 0; nt < 8; ++nt) {
    float bcol = bias[nt * 16 + rc];
#pragma unroll
    for (int r = 0; r < 8; ++r) {
      long o = (long)(wave * 16 + r + 8 * h) * 128 + nt * 16 + rc;
      float v = acc[nt][r];
      XW[o] = v;                         // raw product for edge gather
      HB[o] = bcol + v * dsq[r];         // bias + self-loop init for scatter
    }
  }
}

// ---------------- WMMA GEMM2: [N,128] x [128,8] (N padded to 16) -------------
// Fused epilogue: stores raw z AND out = b2 + z * dis^2.
__global__ void k_gemm2_wmma(const float* __restrict__ H,
                             const float* __restrict__ W2,
                             const float* __restrict__ dis,
                             const float* __restrict__ b2,
                             float* __restrict__ Z,
                             float* __restrict__ OUT, int nStrips) {
  int wave = (blockIdx.x * blockDim.x + threadIdx.x) >> 5;
  if (wave >= nStrips) return;
  int lane = threadIdx.x & 31;
  int rc = lane & 15;
  int h  = lane >> 4;
  long abase = (long)(wave * 16 + rc) * 128;

  v8f acc = (v8f){0,0,0,0,0,0,0,0};
  for (int k0 = 0; k0 < 128; k0 += 4) {
    float2 av = *(const float2*)(H + abase + k0 + 2 * h);
    v2f a; a[0] = av.x; a[1] = av.y;
    v2f b; b[0] = 0.0f; b[1] = 0.0f;
    if (rc < 8) {                            // zero-pad cols 8..15
      b[0] = W2[(k0 + 2 * h)     * 8 + rc];
      b[1] = W2[(k0 + 2 * h + 1) * 8 + rc];
    }
    acc = __builtin_amdgcn_wmma_f32_16x16x4_f32(
        false, a, false, b, (short)0, acc, false, false);
  }
  if (rc < 8) {
    float bcol = b2[rc];
#pragma unroll
    for (int r = 0; r < 8; ++r) {
      int row = wave * 16 + r + 8 * h;
      float t = dis[row];
      long o = (long)row * 8 + rc;
      float v = acc[r];
      Z[o]   = v;
      OUT[o] = bcol + v * (t * t);
    }
  }
}

// scalar tail GEMM (fused epilogue) for row counts not divisible by 16
// (not hit for N=100000)
__global__ void k_gemm_tail(const float* __restrict__ A, const float* __restrict__ W,
                            const float* __restrict__ dis, const float* __restrict__ bias,
                            float* __restrict__ O, float* __restrict__ OB,
                            int row0, int rowEnd, int K, int Ncols) {
  int idx = blockIdx.x * blockDim.x + threadIdx.x;
  int rows = rowEnd - row0;
  if (idx >= rows * Ncols) return;
  int r = row0 + idx / Ncols, c = idx % Ncols;
  float s = 0.0f;
  for (int k = 0; k < K; ++k) s += A[(long)r * K + k] * W[k * Ncols + c];
  long o = (long)r * Ncols + c;
  float t = dis[r];
  O[o]  = s;
  OB[o] = bias[c] + s * (t * t);
}

// ---------------- layer-1 edge scatter + ReLU ----------------
// 32 lanes per edge: one coalesced 512B row gather, 4 f32 atomics per lane.
__global__ void k_edge_agg128(const long long* __restrict__ ei,
                              const float* __restrict__ nrm,
                              const float* __restrict__ xw,
                              float* __restrict__ hbuf, int E) {
  int idx = blockIdx.x * blockDim.x + threadIdx.x;   // E*32
  if (idx >= E * 32) return;
  int e = idx >> 5, q = idx & 31;
  int s = (int)ei[e], d = (int)ei[(long)E + e];
  float w = nrm[e];
  float4 xv = *(const float4*)(xw + (long)s * 128 + q * 4);
  float* hp = hbuf + (long)d * 128 + q * 4;
  atomAddF(hp + 0, xv.x * w);
  atomAddF(hp + 1, xv.y * w);
  atomAddF(hp + 2, xv.z * w);
  atomAddF(hp + 3, xv.w * w);
}

__global__ void k_relu4(float* __restrict__ hbuf, int total4) {
  int idx = blockIdx.x * blockDim.x + threadIdx.x;
  if (idx >= total4) return;
  float4 v = *(float4*)(hbuf + (long)idx * 4);
  v.x = fmaxf(v.x, 0.0f); v.y = fmaxf(v.y, 0.0f);
  v.z = fmaxf(v.z, 0.0f); v.w = fmaxf(v.w, 0.0f);
  *(float4*)(hbuf + (long)idx * 4) = v;
}

// ---------------- layer-2 edge scatter ----------------
__global__ void k_edge_agg8(const long long* __restrict__ ei,
                            const float* __restrict__ nrm,
                            const float* __restrict__ z,
                            float* __restrict__ out, int E) {
  int e = blockIdx.x * blockDim.x + threadIdx.x;
  if (e >= E) return;
  int s = (int)ei[e], d = (int)ei[(long)E + e];
  float w = nrm[e];
  float4 z0 = *(const float4*)(z + (long)s * 8);
  float4 z1 = *(const float4*)(z + (long)s * 8 + 4);
  float* op = out + (long)d * 8;
  atomAddF(op + 0, z0.x * w); atomAddF(op + 1, z0.y * w);
  atomAddF(op + 2, z0.z * w); atomAddF(op + 3, z0.w * w);
  atomAddF(op + 4, z1.x * w); atomAddF(op + 5, z1.y * w);
  atomAddF(op + 6, z1.z * w); atomAddF(op + 7, z1.w * w);
}

__global__ void k_logsoftmax8(float* __restrict__ out, int N) {
  int n = blockIdx.x * blockDim.x + threadIdx.x;
  if (n >= N) return;
  float4 a = *(float4*)(out + (long)n * 8);
  float4 b = *(float4*)(out + (long)n * 8 + 4);
  float v[8] = {a.x, a.y, a.z, a.w, b.x, b.y, b.z, b.w};
  float m = v[0];
#pragma unroll
  for (int i = 1; i < 8; ++i) m = fmaxf(m, v[i]);
  float s = 0.0f;
#pragma unroll
  for (int i = 0; i < 8; ++i) s += expf(v[i] - m);
  float l = logf(s);
  a.x = v[0]-m-l; a.y = v[1]-m-l; a.z = v[2]-m-l; a.w = v[3]-m-l;
  b.x = v[4]-m-l; b.y = v[5]-m-l; b.z = v[6]-m-l; b.w = v[7]-m-l;
  *(float4*)(out + (long)n * 8) = a;
  *(float4*)(out + (long)n * 8 + 4) = b;
}

// ---------------- launch ----------------
extern "C" void kernel_launch(void* const* d_in, const int* in_sizes, int n_in,
                              void* d_out, int out_size, void* d_ws, size_t ws_size,
                              hipStream_t stream) {
  const float*     x  = (const float*)d_in[0];
  const long long* ei = (const long long*)d_in[1];
  const float*     W1 = (const float*)d_in[2];
  const float*     b1 = (const float*)d_in[3];
  const float*     W2 = (const float*)d_in[4];
  const float*     b2 = (const float*)d_in[5];
  float* out = (float*)d_out;

  const int FIN = 128, HD = 128, C = 8;
  int N = in_sizes[0] / FIN;
  int E = in_sizes[1] / 2;

  float* ws   = (float*)d_ws;
  float* xw   = ws;                          // N*128
  float* hbuf = xw   + (size_t)N * HD;       // N*128
  float* z    = hbuf + (size_t)N * HD;       // N*8
  float* dis  = z    + (size_t)N * C;        // N   (deg -> rsqrt in place)
  float* nrm  = dis  + (size_t)N;            // E
  (void)ws_size; (void)n_in; (void)out_size;

  const int B = 256;
  auto G = [](long t, int b) { return (int)((t + b - 1) / b); };

  // normalization
  k_deg_init      <<<G(N, B), B, 0, stream>>>(dis, N);
  k_deg_edges     <<<G(E, B), B, 0, stream>>>(ei, dis, E);
  k_rsqrt_inplace <<<G(N, B), B, 0, stream>>>(dis, N);
  k_edge_norm     <<<G(E, B), B, 0, stream>>>(ei, dis, nrm, E);

  int strips = N / 16;
  int tail   = N - strips * 16;

  // layer 1: WMMA GEMM with fused bias+self-loop init, then edge scatter, ReLU
  k_gemm1_wmma<<<G((long)strips * 32, B), B, 0, stream>>>(x, W1, dis, b1, xw, hbuf, strips);
  if (tail)
    k_gemm_tail<<<G((long)tail * HD, B), B, 0, stream>>>(x, W1, dis, b1, xw, hbuf,
                                                         strips * 16, N, FIN, HD);
  k_edge_agg128<<<G((long)E * 32, B), B, 0, stream>>>(ei, nrm, xw, hbuf, E);
  k_relu4      <<<G((long)N * 32, B), B, 0, stream>>>(hbuf, N * 32);

  // layer 2: WMMA GEMM with fused bias+self-loop init, edge scatter, log-softmax
  k_gemm2_wmma<<<G((long)strips * 32, B), B, 0, stream>>>(hbuf, W2, dis, b2, z, out, strips);
  if (tail)
    k_gemm_tail<<<G((long)tail * C, B), B, 0, stream>>>(hbuf, W2, dis, b2, z, out,
                                                        strips * 16, N, HD, C);
  k_edge_agg8  <<<G(E, B), B, 0, stream>>>(ei, nrm, z, out, E);
  k_logsoftmax8<<<G(N, B), B, 0, stream>>>(out, N);
}